// TreeCRF_50388556316895
// MI455X (gfx1250) — compile-verified
//
#include <hip/hip_runtime.h>
#include <hip/hip_bf16.h>
#include <stdint.h>

typedef __attribute__((ext_vector_type(2))) float v2f;
typedef __attribute__((ext_vector_type(8))) float v8f;

#define B_DIM   128
#define NNODES  2047   // 2*1024-1
#define D_INK   512
#define NL      64

// ---------------------------------------------------------------------------
// Async global->LDS copy of one contiguous 16B-granular region.
// Each call copies 16 bytes (one b128) per calling thread.
// Caller must s_wait_asynccnt + barrier before reading the LDS.
// ---------------------------------------------------------------------------
__device__ __forceinline__ void async_copy_b128(unsigned lds_byte_off,
                                                const void* gsrc)
{
    asm volatile("global_load_async_to_lds_b128 %0, %1, off"
                 :: "v"(lds_byte_off), "v"(gsrc) : "memory");
}
__device__ __forceinline__ void async_wait_all()
{
    asm volatile("s_wait_asynccnt 0x0" ::: "memory");
}

// ---------------------------------------------------------------------------
// Kernel 1: emis = hidden @ W + b   (fp32 WMMA 16x16x4)
// grid = 2047 blocks * 8 waves * 16 rows = 262,016 rows = B*NNODES
// Each wave: 16 rows x all 64 cols, K=512. W staged in LDS per 128-K chunk
// via GLOBAL_LOAD_ASYNC_TO_LDS_B128 (TDM-style async path, ASYNCcnt).
// ---------------------------------------------------------------------------
__global__ __launch_bounds__(256) void emis_kernel(
    const float* __restrict__ hidden, const float* __restrict__ W,
    const float* __restrict__ bias, float* __restrict__ emis)
{
    __shared__ float sW[128 * 64];          // 32 KB chunk of W (row-major)

    const int tid  = threadIdx.x;
    const int lane = tid & 31;
    const int wave = tid >> 5;
    const int nn   = lane & 15;             // tile row (A) / tile col (B)
    const int half = lane >> 4;             // K-half select per ISA fp32 layout

    const size_t rowBase = ((size_t)blockIdx.x * 8 + wave) * 16;
    const float* aRow = hidden + (rowBase + (size_t)nn) * D_INK + 2 * half;

    const unsigned ldsW = (unsigned)(uintptr_t)sW;   // LDS byte offset of sW

    v8f acc[4] = {};                        // 4 col-tiles of 16

    for (int c = 0; c < 4; ++c) {
        __syncthreads();                    // protect LDS before overwrite
        // async DMA: 32 KB contiguous W chunk -> LDS (8 x b128 per thread)
        {
            const char* g = (const char*)(W + (size_t)c * 128 * NL) + tid * 16;
            #pragma unroll
            for (int i = 0; i < 8; ++i)
                async_copy_b128(ldsW + tid * 16 + i * 4096, g + i * 4096);
        }
        if (c < 3) __builtin_prefetch(aRow + (c + 1) * 128, 0, 0);
        async_wait_all();
        __syncthreads();

        #pragma unroll 4
        for (int k0 = 0; k0 < 128; k0 += 4) {
            // A fragment: lane<16 -> K={k0,k0+1}, lane>=16 -> K={k0+2,k0+3}
            float2 a2 = *(const float2*)(aRow + c * 128 + k0);
            v2f a; a.x = a2.x; a.y = a2.y;
            const int kk = k0 + 2 * half;
            #pragma unroll
            for (int t = 0; t < 4; ++t) {
                v2f b;
                b.x = sW[kk * 64 + t * 16 + nn];
                b.y = sW[(kk + 1) * 64 + t * 16 + nn];
                acc[t] = __builtin_amdgcn_wmma_f32_16x16x4_f32(
                    false, a, false, b, (short)0, acc[t], false, false);
            }
        }
    }

    // store D (16x16 f32 layout: lane<16 rows 0..7, lane>=16 rows 8..15) + bias
    #pragma unroll
    for (int t = 0; t < 4; ++t) {
        float bv = bias[t * 16 + nn];
        #pragma unroll
        for (int r = 0; r < 8; ++r) {
            size_t row = rowBase + (size_t)(half * 8 + r);
            emis[row * NL + t * 16 + nn] = acc[t][r] + bv;
        }
    }
}

// ---------------------------------------------------------------------------
// Kernel 2: expT = exp(trans)  (64x64)
// ---------------------------------------------------------------------------
__global__ void exp_kernel(const float* __restrict__ t, float* __restrict__ o)
{
    int i = blockIdx.x * blockDim.x + threadIdx.x;
    if (i < NL * NL) o[i] = __expf(t[i]);
}

// ---------------------------------------------------------------------------
// Kernel 3: one tree level.
// ins[b,p,:] += log(expT @ exp(L-mL)) + mL + log(expT @ exp(R-mR)) + mR
// block = 128 threads (4 waves); wave = 16 parents; lane = one child vector.
// grid = 2*n blocks (128*n parents / 64 per block, exact).
// expT A-fragments read straight from global (16 KB, L2/L0 resident).
// sEV pair-interleaved so B fragments are single aligned ds_load_b64.
// ---------------------------------------------------------------------------
__global__ __launch_bounds__(128) void level_kernel(
    const float* __restrict__ expT, float* __restrict__ ins,
    int n, int dshift)
{
    __shared__ float2 sEV[4][32][33];       // [wave][k-pair][col], ~34 KB
    __shared__ float  sMax[4][32];

    const int tid  = threadIdx.x;
    const int lane = tid & 31;
    const int wave = tid >> 5;
    const int nn   = lane & 15;
    const int half = lane >> 4;

    // ---- phase 1: each lane owns one child vector (16 parents x {L,R}) ----
    {
        int f = (blockIdx.x * 4 + wave) * 16 + nn;   // parent flat id in [0,128n)
        int b = f >> dshift;
        int i = f & (n - 1);
        int p = n - 1 + i;
        int child = 2 * p + 1 + half;                // half=0 -> left, 1 -> right
        const float* v = ins + ((size_t)b * NNODES + child) * NL;

        float4 buf[16];
        float mx = -3.402823466e38f;
        #pragma unroll
        for (int m4 = 0; m4 < 16; ++m4) {
            buf[m4] = *(const float4*)(v + m4 * 4);
            mx = fmaxf(mx, fmaxf(fmaxf(buf[m4].x, buf[m4].y),
                                 fmaxf(buf[m4].z, buf[m4].w)));
        }
        sMax[wave][lane] = mx;
        #pragma unroll
        for (int m4 = 0; m4 < 16; ++m4) {
            sEV[wave][m4 * 2 + 0][lane] =
                make_float2(__expf(buf[m4].x - mx), __expf(buf[m4].y - mx));
            sEV[wave][m4 * 2 + 1][lane] =
                make_float2(__expf(buf[m4].z - mx), __expf(buf[m4].w - mx));
        }
    }
    __syncthreads();

    // ---- phase 2: U_L = expT @ EV_L, U_R = expT @ EV_R via fp32 WMMA ----
    v8f accL[4] = {};
    v8f accR[4] = {};
    #pragma unroll 2
    for (int k0 = 0; k0 < NL; k0 += 4) {
        const int kk = k0 + 2 * half;                // even
        float2 l2 = sEV[wave][kk >> 1][nn];
        float2 r2 = sEV[wave][kk >> 1][16 + nn];
        v2f bL; bL.x = l2.x; bL.y = l2.y;
        v2f bR; bR.x = r2.x; bR.y = r2.y;
        #pragma unroll
        for (int t = 0; t < 4; ++t) {
            float2 a2 = *(const float2*)(expT + (t * 16 + nn) * NL + kk);
            v2f a; a.x = a2.x; a.y = a2.y;
            accL[t] = __builtin_amdgcn_wmma_f32_16x16x4_f32(
                false, a, false, bL, (short)0, accL[t], false, false);
            accR[t] = __builtin_amdgcn_wmma_f32_16x16x4_f32(
                false, a, false, bR, (short)0, accR[t], false, false);
        }
    }

    // ---- phase 3: ins[b,p,l] = emis + log(uL)+mL + log(uR)+mR ----
    {
        int f = (blockIdx.x * 4 + wave) * 16 + nn;
        int b = f >> dshift;
        int i = f & (n - 1);
        int p = n - 1 + i;
        float mL = sMax[wave][nn];
        float mR = sMax[wave][16 + nn];
        float* dst = ins + ((size_t)b * NNODES + p) * NL;
        #pragma unroll
        for (int t = 0; t < 4; ++t) {
            #pragma unroll
            for (int r = 0; r < 8; ++r) {
                int l = t * 16 + half * 8 + r;
                dst[l] = dst[l] + __logf(accL[t][r]) + mL
                                + __logf(accR[t][r]) + mR;
            }
        }
    }
}

// ---------------------------------------------------------------------------
// Kernel 4: copy root inside -> output (128 x 64)
// ---------------------------------------------------------------------------
__global__ void out_kernel(const float* __restrict__ ins, float* __restrict__ out)
{
    int i = blockIdx.x * blockDim.x + threadIdx.x;
    if (i < B_DIM * NL) {
        int b = i >> 6, l = i & 63;
        out[i] = ins[(size_t)b * NNODES * NL + l];
    }
}

// ---------------------------------------------------------------------------
extern "C" void kernel_launch(void* const* d_in, const int* in_sizes, int n_in,
                              void* d_out, int out_size, void* d_ws, size_t ws_size,
                              hipStream_t stream)
{
    const float* hidden = (const float*)d_in[0];  // (128, 2047, 512)
    const float* W      = (const float*)d_in[1];  // (512, 64)
    const float* bias   = (const float*)d_in[2];  // (64,)
    const float* trans  = (const float*)d_in[3];  // (64, 64)

    float* ins  = (float*)d_ws;                               // B*NNODES*NL fp32
    float* expT = ins + (size_t)B_DIM * NNODES * NL;          // 64*64 fp32

    // 1) emis into 'ins' (leaves are already their inside values)
    emis_kernel<<<2047, 256, 0, stream>>>(hidden, W, bias, ins);
    // 2) exp(trans)
    exp_kernel<<<16, 256, 0, stream>>>(trans, expT);
    // 3) bottom-up tree levels: d = 9..0, n = 2^d parents per batch
    for (int d = 9; d >= 0; --d) {
        int nlev = 1 << d;
        level_kernel<<<2 * nlev, 128, 0, stream>>>(expT, ins, nlev, d);
    }
    // 4) root -> output
    out_kernel<<<32, 256, 0, stream>>>(ins, (float*)d_out);
}